// PointTransformerLayer_MH_85959475462564
// MI455X (gfx1250) — compile-verified
//
#include <hip/hip_runtime.h>
#include <hip/hip_bf16.h>

#define N_PTS 30000
#define C_IN  128
#define O_OUT 128
#define H_HEADS 8
#define S_NBR 16
#define D_DIM 16
#define OS_DIM 16
#define EPSF 1e-5f

typedef __attribute__((ext_vector_type(2))) float v2f;
typedef __attribute__((ext_vector_type(8))) float v8f;

// D = A(16x4) * B(4x16) + C, f32 WMMA.
// A layout (per ISA): lane L holds row m=L&15, K pair {2*(L>>4), 2*(L>>4)+1}.
// B layout (mirrors C pattern): VGPR g holds K row g (lanes 0-15) / g+2 (lanes 16-31).
// C/D layout: lane L holds col n=L&15, rows g + 8*(L>>4) for g in 0..7.
__device__ inline v8f wmma_f32(v2f a, v2f b, v8f c) {
    return __builtin_amdgcn_wmma_f32_16x16x4_f32(false, a, false, b, (short)0, c, false, false);
}

// --- CDNA5 async global->LDS gather (ASYNCcnt-tracked), with safe fallback --
#if __has_builtin(__builtin_amdgcn_global_load_async_to_lds_b128)
#define HAVE_ASYNC_LDS 1
typedef int v4i_vs __attribute__((vector_size(16)));            // matches builtin's pointee
typedef __attribute__((address_space(1))) v4i_vs* g_v4i_p;      // global (AS1)
typedef __attribute__((address_space(3))) v4i_vs* l_v4i_p;      // LDS (AS3)
__device__ inline void async_copy16(const float* g, float* l) {
    __builtin_amdgcn_global_load_async_to_lds_b128((g_v4i_p)g, (l_v4i_p)l, 0, 0);
}
__device__ inline void wait_async0() {
#if __has_builtin(__builtin_amdgcn_s_wait_asynccnt)
    __builtin_amdgcn_s_wait_asynccnt(0);
#else
    asm volatile("s_wait_asynccnt 0x0" ::: "memory");
#endif
}
#endif

// ---------------------------------------------------------------------------
// Kernel 1: [xq|xk|xv] = x @ W{q,k,v} + b{q,k,v}.  One wave per 16x16 tile.
// grid = (N/16, 3), block = 256 (8 waves -> 8 column tiles of one row tile).
// ---------------------------------------------------------------------------
__global__ __launch_bounds__(256)
void qkv_wmma(const float* __restrict__ x,
              const float* __restrict__ Wq, const float* __restrict__ bq,
              const float* __restrict__ Wk, const float* __restrict__ bk,
              const float* __restrict__ Wv, const float* __restrict__ bv,
              float* __restrict__ xq, float* __restrict__ xk, float* __restrict__ xv)
{
    const int rowTile = blockIdx.x;           // 0..1874 (30000/16)
    const int which   = blockIdx.y;           // 0=q 1=k 2=v
    const float* W; const float* bias; float* out;
    if (which == 0)      { W = Wq; bias = bq; out = xq; }
    else if (which == 1) { W = Wk; bias = bk; out = xk; }
    else                 { W = Wv; bias = bv; out = xv; }

    const int wave = threadIdx.x >> 5;        // 0..7 -> column tile
    const int lane = threadIdx.x & 31;
    const int m16  = lane & 15;
    const int hi   = lane >> 4;
    const int row  = rowTile * 16 + m16;      // A row for this lane
    const int n    = wave * 16 + m16;         // B/C column for this lane

    v8f c = {};
    const float* arow = x + (size_t)row * C_IN;
#pragma unroll
    for (int k0 = 0; k0 < C_IN; k0 += 4) {
        const int kk = k0 + 2 * hi;
        v2f a, b;
        a.x = arow[kk];
        a.y = arow[kk + 1];
        b.x = W[(size_t)(kk + 0) * O_OUT + n];
        b.y = W[(size_t)(kk + 1) * O_OUT + n];
        c = wmma_f32(a, b, c);
    }
    const float bn = bias[n];
#pragma unroll
    for (int g = 0; g < 8; ++g) {
        const int r = rowTile * 16 + g + 8 * hi;
        out[(size_t)r * O_OUT + n] = c[g] + bn;
    }
}

// ---------------------------------------------------------------------------
// Kernel 2: fused per-point attention.  One 128-thread block (4 waves)/point.
// Gathered xk/xv rows are moved global->LDS via the async engine and the
// latency is overlapped with the positional MLP + pe WMMAs.
// ---------------------------------------------------------------------------
__global__ __launch_bounds__(128)
void pt_fused(const float* __restrict__ p, const int* __restrict__ idx,
              const float* __restrict__ xq, const float* __restrict__ xk,
              const float* __restrict__ xv,
              const float* __restrict__ Wp1, const float* __restrict__ bp1,
              const float* __restrict__ gp,  const float* __restrict__ betap,
              const float* __restrict__ Wp2, const float* __restrict__ bp2,
              const float* __restrict__ gw1, const float* __restrict__ betaw1,
              const float* __restrict__ Ww1, const float* __restrict__ bw1,
              const float* __restrict__ gw2, const float* __restrict__ betaw2,
              const float* __restrict__ Ww2, const float* __restrict__ bw2,
              float* __restrict__ outp)
{
    const int i    = blockIdx.x;              // point index
    const int t    = threadIdx.x;             // 0..127
    const int wave = t >> 5;                  // 0..3
    const int lane = t & 31;
    const int m16  = lane & 15;
    const int hi   = lane >> 4;

    __shared__ int   s_idx[S_NBR];
    __shared__ float s_pi[4];
    __shared__ float s_t[S_NBR][4];           // LN'd+relu'd 3-vec, padded to 4
    __shared__ float s_kg[S_NBR][O_OUT];      // gathered xk rows (async)
    __shared__ float s_vg[S_NBR][O_OUT];      // gathered xv rows (async)
    __shared__ float s_pe[S_NBR][O_OUT];      // positional embedding (kept to end)
    __shared__ float s_r[S_NBR][O_OUT];       // r = kg + pe - xq
    __shared__ float s_ln[S_NBR * H_HEADS][D_DIM];   // per-(s,h) LN'd rows (A operand)
    __shared__ float s_w[S_NBR * H_HEADS][OS_DIM];   // WMMA outputs
    __shared__ float s_logit[S_NBR][H_HEADS];
    __shared__ float s_wgt[S_NBR][H_HEADS];

    if (t < S_NBR) s_idx[t] = idx[(size_t)i * S_NBR + t];
    if (t < 3)     s_pi[t]  = p[(size_t)i * 3 + t];
    __syncthreads();

    // --- kick off the 32 row gathers (16 xk + 16 xv, 512B each) -----------
    // Each lane moves 16B; one wave covers one row; 4 waves split the 16 s.
#ifdef HAVE_ASYNC_LDS
    for (int s = wave; s < S_NBR; s += 4) {
        const size_t base = (size_t)s_idx[s] * O_OUT + lane * 4;
        async_copy16(xk + base, &s_kg[s][lane * 4]);
        async_copy16(xv + base, &s_vg[s][lane * 4]);
    }
#else
    for (int s = wave; s < S_NBR; s += 4) {
        const size_t base = (size_t)s_idx[s] * O_OUT + lane * 4;
        *(float4*)&s_kg[s][lane * 4] = *(const float4*)(xk + base);
        *(float4*)&s_vg[s][lane * 4] = *(const float4*)(xv + base);
    }
#endif

    // --- positional MLP stage 1: (pr @ Wp1 + bp1) -> LN(3) -> relu ---------
    if (t < S_NBR) {
        const int j = s_idx[t];
        float pr0 = p[(size_t)j * 3 + 0] - s_pi[0];
        float pr1 = p[(size_t)j * 3 + 1] - s_pi[1];
        float pr2 = p[(size_t)j * 3 + 2] - s_pi[2];
        float u[3];
#pragma unroll
        for (int o = 0; o < 3; ++o)
            u[o] = pr0 * Wp1[0 * 3 + o] + pr1 * Wp1[1 * 3 + o] + pr2 * Wp1[2 * 3 + o] + bp1[o];
        float mu = (u[0] + u[1] + u[2]) * (1.0f / 3.0f);
        float v = 0.f;
#pragma unroll
        for (int o = 0; o < 3; ++o) { float d = u[o] - mu; v += d * d; }
        v *= (1.0f / 3.0f);
        float rinv = rsqrtf(v + EPSF);
#pragma unroll
        for (int o = 0; o < 3; ++o) {
            float z = (u[o] - mu) * rinv * gp[o] + betap[o];
            s_t[t][o] = fmaxf(z, 0.0f);
        }
        s_t[t][3] = 0.0f;                      // K pad for WMMA (kills row-3 term)
    }
    __syncthreads();

    // --- pe = t @ Wp2 + bp2 via WMMA (K=4, rows = s).  2 col tiles / wave. --
#pragma unroll
    for (int ct = wave; ct < 8; ct += 4) {
        const int n = ct * 16 + m16;
        const int kk = 2 * hi;
        v2f a, b;
        a.x = s_t[m16][kk];
        a.y = s_t[m16][kk + 1];                // == 0 for hi==1 (pad)
        b.x = Wp2[(size_t)kk * O_OUT + n];     // rows 0,2
        b.y = (hi == 0) ? Wp2[(size_t)1 * O_OUT + n] : 0.0f;  // row 1 / padded row 3
        v8f c = {};
        c = wmma_f32(a, b, c);
        const float bn = bp2[n];
#pragma unroll
        for (int g = 0; g < 8; ++g)
            s_pe[g + 8 * hi][n] = c[g] + bn;
    }

    // gathers must have landed before r-stage; pe barrier doubles as LDS fence
#ifdef HAVE_ASYNC_LDS
    wait_async0();
#endif
    __syncthreads();

    // --- r = kg + pe - xq  (thread t owns channel o = t) -------------------
    {
        const float q = xq[(size_t)i * O_OUT + t];
#pragma unroll
        for (int s = 0; s < S_NBR; ++s)
            s_r[s][t] = s_kg[s][t] + s_pe[s][t] - q;
    }
    __syncthreads();

    // --- LN over D=16 per (s,h), relu -> s_ln row (row id = s*8+h = t) -----
    {
        const int s = t >> 3, h = t & 7;
        float acc = 0.f;
#pragma unroll
        for (int d = 0; d < D_DIM; ++d) acc += s_r[s][h * 16 + d];
        const float mu = acc * (1.0f / 16.0f);
        float v = 0.f;
#pragma unroll
        for (int d = 0; d < D_DIM; ++d) { float dd = s_r[s][h * 16 + d] - mu; v += dd * dd; }
        const float rinv = rsqrtf(v * (1.0f / 16.0f) + EPSF);
#pragma unroll
        for (int d = 0; d < D_DIM; ++d) {
            float z = (s_r[s][h * 16 + d] - mu) * rinv * gw1[d] + betaw1[d];
            s_ln[t][d] = fmaxf(z, 0.0f);
        }
    }
    __syncthreads();

    // --- w1 = ln @ Ww1 + bw1 via WMMA (8 row tiles of 16 (s,h) rows) -------
#pragma unroll
    for (int rt = wave; rt < 8; rt += 4) {
        v8f c = {};
#pragma unroll
        for (int k0 = 0; k0 < D_DIM; k0 += 4) {
            const int kk = k0 + 2 * hi;
            const int mrow = rt * 16 + m16;
            v2f a, b;
            a.x = s_ln[mrow][kk];
            a.y = s_ln[mrow][kk + 1];
            b.x = Ww1[(kk + 0) * OS_DIM + m16];
            b.y = Ww1[(kk + 1) * OS_DIM + m16];
            c = wmma_f32(a, b, c);
        }
        const float bn = bw1[m16];
#pragma unroll
        for (int g = 0; g < 8; ++g)
            s_w[rt * 16 + g + 8 * hi][m16] = c[g] + bn;
    }
    __syncthreads();

    // --- LN over OS=16 per row, relu -> s_ln -------------------------------
    {
        float acc = 0.f;
#pragma unroll
        for (int d = 0; d < OS_DIM; ++d) acc += s_w[t][d];
        const float mu = acc * (1.0f / 16.0f);
        float v = 0.f;
#pragma unroll
        for (int d = 0; d < OS_DIM; ++d) { float dd = s_w[t][d] - mu; v += dd * dd; }
        const float rinv = rsqrtf(v * (1.0f / 16.0f) + EPSF);
#pragma unroll
        for (int d = 0; d < OS_DIM; ++d) {
            float z = (s_w[t][d] - mu) * rinv * gw2[d] + betaw2[d];
            s_ln[t][d] = fmaxf(z, 0.0f);
        }
    }
    __syncthreads();

    // --- w2 = ln @ Ww2 + bw2 via WMMA --------------------------------------
#pragma unroll
    for (int rt = wave; rt < 8; rt += 4) {
        v8f c = {};
#pragma unroll
        for (int k0 = 0; k0 < OS_DIM; k0 += 4) {
            const int kk = k0 + 2 * hi;
            const int mrow = rt * 16 + m16;
            v2f a, b;
            a.x = s_ln[mrow][kk];
            a.y = s_ln[mrow][kk + 1];
            b.x = Ww2[(kk + 0) * OS_DIM + m16];
            b.y = Ww2[(kk + 1) * OS_DIM + m16];
            c = wmma_f32(a, b, c);
        }
        const float bn = bw2[m16];
#pragma unroll
        for (int g = 0; g < 8; ++g)
            s_w[rt * 16 + g + 8 * hi][m16] = c[g] + bn;
    }
    __syncthreads();

    // --- logit = mean over OS ----------------------------------------------
    {
        float acc = 0.f;
#pragma unroll
        for (int d = 0; d < OS_DIM; ++d) acc += s_w[t][d];
        s_logit[t >> 3][t & 7] = acc * (1.0f / 16.0f);
    }
    __syncthreads();

    // --- softmax over S per head -------------------------------------------
    if (t < H_HEADS) {
        float mx = -1e30f;
#pragma unroll
        for (int s = 0; s < S_NBR; ++s) mx = fmaxf(mx, s_logit[s][t]);
        float sum = 0.f;
#pragma unroll
        for (int s = 0; s < S_NBR; ++s) { float e = __expf(s_logit[s][t] - mx); s_wgt[s][t] = e; sum += e; }
        const float inv = 1.0f / sum;
#pragma unroll
        for (int s = 0; s < S_NBR; ++s) s_wgt[s][t] *= inv;
    }
    __syncthreads();

    // --- out[i,o] = sum_s (vg + pe) * w  (vg already in LDS) ---------------
    {
        const int h = t >> 4;
        float acc = 0.f;
#pragma unroll
        for (int s = 0; s < S_NBR; ++s)
            acc += (s_vg[s][t] + s_pe[s][t]) * s_wgt[s][h];
        outp[(size_t)i * O_OUT + t] = acc;
    }
}

extern "C" void kernel_launch(void* const* d_in, const int* in_sizes, int n_in,
                              void* d_out, int out_size, void* d_ws, size_t ws_size,
                              hipStream_t stream) {
    const float* p     = (const float*)d_in[0];
    const float* x     = (const float*)d_in[1];
    const int*   idx   = (const int*)  d_in[2];
    const float* Wq    = (const float*)d_in[3];
    const float* bq    = (const float*)d_in[4];
    const float* Wk    = (const float*)d_in[5];
    const float* bk    = (const float*)d_in[6];
    const float* Wv    = (const float*)d_in[7];
    const float* bv    = (const float*)d_in[8];
    const float* Wp1   = (const float*)d_in[9];
    const float* bp1   = (const float*)d_in[10];
    const float* gp    = (const float*)d_in[11];
    const float* betap = (const float*)d_in[12];
    const float* Wp2   = (const float*)d_in[13];
    const float* bp2   = (const float*)d_in[14];
    const float* gw1   = (const float*)d_in[15];
    const float* betaw1= (const float*)d_in[16];
    const float* Ww1   = (const float*)d_in[17];
    const float* bw1   = (const float*)d_in[18];
    const float* gw2   = (const float*)d_in[19];
    const float* betaw2= (const float*)d_in[20];
    const float* Ww2   = (const float*)d_in[21];
    const float* bw2   = (const float*)d_in[22];

    float* xq = (float*)d_ws;
    float* xk = xq + (size_t)N_PTS * O_OUT;
    float* xv = xk + (size_t)N_PTS * O_OUT;

    qkv_wmma<<<dim3(N_PTS / 16, 3), 256, 0, stream>>>(x, Wq, bq, Wk, bk, Wv, bv, xq, xk, xv);

    pt_fused<<<dim3(N_PTS), 128, 0, stream>>>(p, idx, xq, xk, xv,
                                              Wp1, bp1, gp, betap, Wp2, bp2,
                                              gw1, betaw1, Ww1, bw1,
                                              gw2, betaw2, Ww2, bw2,
                                              (float*)d_out);
}